// GNNGraphColoring_39711267619177
// MI455X (gfx1250) — compile-verified
//
#include <hip/hip_runtime.h>
#include <hip/hip_bf16.h>

typedef __attribute__((ext_vector_type(2))) float v2f;
typedef __attribute__((ext_vector_type(8))) float v8f;

#define FDIM 64

// ---------------------------------------------------------------------------
// GEMM: C[M x 64] = A[M x 64] * W[64 x 64] using v_wmma_f32_16x16x4_f32.
// One block = 128 threads = 4 waves; block handles one 16-row tile, wave w
// handles column tile w (16 cols). A tile (16x64) and full W (64x64) staged
// in LDS. M must be a multiple of 16 (50000 = 3125*16).
// ---------------------------------------------------------------------------
__global__ void __launch_bounds__(128)
gcn_gemm64_wmma(const float* __restrict__ A, const float* __restrict__ W,
                float* __restrict__ C) {
  __shared__ float lA[16 * FDIM];     //  4 KB
  __shared__ float lW[FDIM * FDIM];   // 16 KB
  const int tid = threadIdx.x;
  const int rowbase = blockIdx.x * 16;

  for (int i = tid; i < 16 * FDIM; i += 128) lA[i] = A[(size_t)rowbase * FDIM + i];
  for (int i = tid; i < FDIM * FDIM; i += 128) lW[i] = W[i];
  __syncthreads();

  const int wave = tid >> 5;       // column tile 0..3
  const int lane = tid & 31;
  const int half = lane >> 4;      // 0: lanes 0-15, 1: lanes 16-31
  const int mr   = lane & 15;      // M index for A, N index for B/C
  const int ncol = wave * 16 + mr;

  v8f acc = {};
#pragma unroll
  for (int k0 = 0; k0 < FDIM; k0 += 4) {
    const int ka = k0 + 2 * half;
    v2f a, b;
    a.x = lA[mr * FDIM + ka];          // A: 16x4 layout, v0 <-> K=2*half
    a.y = lA[mr * FDIM + ka + 1];      //               v1 <-> K=2*half+1
    b.x = lW[ka * FDIM + ncol];        // B: 4x16, row striped across lanes
    b.y = lW[(ka + 1) * FDIM + ncol];
    acc = __builtin_amdgcn_wmma_f32_16x16x4_f32(false, a, false, b,
                                                (short)0, acc, false, false);
  }
#pragma unroll
  for (int r = 0; r < 8; ++r) {
    const int row = rowbase + r + 8 * half;   // C/D layout: VGPR r <-> M=r(+8)
    C[(size_t)row * FDIM + ncol] = acc[r];
  }
}

// ---------------------------------------------------------------------------
// Final-layer GEMM: C[M x 16] = A[M x 64] * W2pad[64 x 16], where W2 is
// 64x10 row-major and columns 10..15 are zero-padded. One wave per block.
// ---------------------------------------------------------------------------
__global__ void __launch_bounds__(32)
gcn_gemm_cls_wmma(const float* __restrict__ A, const float* __restrict__ W2,
                  float* __restrict__ C) {
  const int rowbase = blockIdx.x * 16;
  const int lane = threadIdx.x & 31;
  const int half = lane >> 4;
  const int mr   = lane & 15;

  v8f acc = {};
#pragma unroll
  for (int k0 = 0; k0 < FDIM; k0 += 4) {
    const int ka = k0 + 2 * half;
    v2f a, b;
    a.x = A[(size_t)(rowbase + mr) * FDIM + ka];
    a.y = A[(size_t)(rowbase + mr) * FDIM + ka + 1];
    b.x = (mr < 10) ? W2[ka * 10 + mr] : 0.0f;
    b.y = (mr < 10) ? W2[(ka + 1) * 10 + mr] : 0.0f;
    acc = __builtin_amdgcn_wmma_f32_16x16x4_f32(false, a, false, b,
                                                (short)0, acc, false, false);
  }
#pragma unroll
  for (int r = 0; r < 8; ++r) {
    const int row = rowbase + r + 8 * half;
    C[(size_t)row * 16 + mr] = acc[r];
  }
}

// ---------------------------------------------------------------------------
// Edge scatter: out[dst[e]*D + f] += H[src[e]*D + f] * ew[e].
// Thread = (edge, feature). D = 64: a wave covers 32 contiguous features of
// one edge -> wave-uniform edge loads, coalesced 128B gather/atomic segments.
// ---------------------------------------------------------------------------
__global__ void __launch_bounds__(256)
gcn_scatter64(const float* __restrict__ H, const long long* __restrict__ src,
              const long long* __restrict__ dst, const float* __restrict__ ew,
              float* __restrict__ out, long long total) {
  const long long t = (long long)blockIdx.x * 256 + threadIdx.x;
  if (t >= total) return;
  const int e = (int)(t >> 6);
  const int f = (int)(t & 63);
  const int s = (int)src[e];
  const int d = (int)dst[e];
  const float w = ew[e];
  atomicAdd(&out[(size_t)d * FDIM + f], H[(size_t)s * FDIM + f] * w);
}

__global__ void __launch_bounds__(256)
gcn_scatter16(const float* __restrict__ H, const long long* __restrict__ src,
              const long long* __restrict__ dst, const float* __restrict__ ew,
              float* __restrict__ out, long long total) {
  const long long t = (long long)blockIdx.x * 256 + threadIdx.x;
  if (t >= total) return;
  const int e = (int)(t >> 4);
  const int f = (int)(t & 15);
  const int s = (int)src[e];
  const int d = (int)dst[e];
  const float w = ew[e];
  atomicAdd(&out[(size_t)d * 16 + f], H[(size_t)s * 16 + f] * w);
}

// ---------------------------------------------------------------------------
// Bias + softmax over 64 features: one wave per node, 2 features per lane,
// wave32 shuffle reductions.
// ---------------------------------------------------------------------------
__global__ void __launch_bounds__(256)
gcn_bias_softmax64(const float* __restrict__ in, const float* __restrict__ bias,
                   float* __restrict__ out, int nNodes) {
  const int node = (blockIdx.x * 256 + threadIdx.x) >> 5;
  const int lane = threadIdx.x & 31;
  if (node >= nNodes) return;
  const size_t base = (size_t)node * FDIM;
  float v0 = in[base + lane] + bias[lane];
  float v1 = in[base + lane + 32] + bias[lane + 32];
  float m = fmaxf(v0, v1);
#pragma unroll
  for (int off = 16; off; off >>= 1) m = fmaxf(m, __shfl_xor(m, off, 32));
  const float e0 = __expf(v0 - m);
  const float e1 = __expf(v1 - m);
  float s = e0 + e1;
#pragma unroll
  for (int off = 16; off; off >>= 1) s += __shfl_xor(s, off, 32);
  const float inv = __frcp_rn(s);
  out[base + lane]      = e0 * inv;
  out[base + lane + 32] = e1 * inv;
}

// Final bias + softmax over 10 classes (input padded to stride 16).
__global__ void __launch_bounds__(256)
gcn_bias_softmax_cls(const float* __restrict__ in, const float* __restrict__ b2,
                     float* __restrict__ out, int nNodes) {
  const int n = blockIdx.x * 256 + threadIdx.x;
  if (n >= nNodes) return;
  float v[10];
  float m = -3.4e38f;
#pragma unroll
  for (int i = 0; i < 10; ++i) {
    v[i] = in[(size_t)n * 16 + i] + b2[i];
    m = fmaxf(m, v[i]);
  }
  float s = 0.0f;
#pragma unroll
  for (int i = 0; i < 10; ++i) {
    v[i] = __expf(v[i] - m);
    s += v[i];
  }
  const float inv = __frcp_rn(s);
#pragma unroll
  for (int i = 0; i < 10; ++i) out[(size_t)n * 10 + i] = v[i] * inv;
}

// ---------------------------------------------------------------------------
// Launcher
// ---------------------------------------------------------------------------
extern "C" void kernel_launch(void* const* d_in, const int* in_sizes, int n_in,
                              void* d_out, int out_size, void* d_ws, size_t ws_size,
                              hipStream_t stream) {
  const float*     x  = (const float*)d_in[0];           // [N, 64]
  const long long* ei = (const long long*)d_in[1];       // [2, E] int64
  const float*     ew = (const float*)d_in[2];           // [E]
  const float*     W0 = (const float*)d_in[3];
  const float*     b0 = (const float*)d_in[4];
  const float*     W1 = (const float*)d_in[5];
  const float*     b1 = (const float*)d_in[6];
  const float*     W2 = (const float*)d_in[7];           // [64, 10]
  const float*     b2 = (const float*)d_in[8];

  const int N = in_sizes[0] / FDIM;     // 50000 (multiple of 16)
  const int E = in_sizes[2];            // 800000
  const long long* src = ei;
  const long long* dst = ei + E;

  float* bufA = (float*)d_ws;                 // [N,64]
  float* bufB = bufA + (size_t)N * FDIM;      // [N,64]
  float* bufC = bufB + (size_t)N * FDIM;      // [N,16]
  float* bufD = bufC + (size_t)N * 16;        // [N,16]

  const int rowTiles = N / 16;                              // 3125
  const long long ef64 = (long long)E * 64;
  const long long ef16 = (long long)E * 16;
  const int scat64Blocks = (int)((ef64 + 255) / 256);
  const int scat16Blocks = (int)((ef16 + 255) / 256);
  const int smaxBlocks   = (N * 32 + 255) / 256;            // wave per node
  const int clsBlocks    = (N + 255) / 256;

  // ---- Layer 0: t0 = x @ W0 ; agg ; +b0 ; softmax -> h1 (bufA) ----
  gcn_gemm64_wmma<<<rowTiles, 128, 0, stream>>>(x, W0, bufA);
  hipMemsetAsync(bufB, 0, (size_t)N * FDIM * sizeof(float), stream);
  gcn_scatter64<<<scat64Blocks, 256, 0, stream>>>(bufA, src, dst, ew, bufB, ef64);
  gcn_bias_softmax64<<<smaxBlocks, 256, 0, stream>>>(bufB, b0, bufA, N);

  // ---- Layer 1: t1 = h1 @ W1 ; agg ; +b1 ; softmax -> h2 (bufB) ----
  gcn_gemm64_wmma<<<rowTiles, 128, 0, stream>>>(bufA, W1, bufB);
  hipMemsetAsync(bufA, 0, (size_t)N * FDIM * sizeof(float), stream);
  gcn_scatter64<<<scat64Blocks, 256, 0, stream>>>(bufB, src, dst, ew, bufA, ef64);
  gcn_bias_softmax64<<<smaxBlocks, 256, 0, stream>>>(bufA, b1, bufB, N);

  // ---- Layer 2: t2 = h2 @ W2 (padded to 16 cols) ; agg ; +b2 ; softmax ----
  gcn_gemm_cls_wmma<<<rowTiles, 32, 0, stream>>>(bufB, W2, bufC);
  hipMemsetAsync(bufD, 0, (size_t)N * 16 * sizeof(float), stream);
  gcn_scatter16<<<scat16Blocks, 256, 0, stream>>>(bufC, src, dst, ew, bufD, ef16);
  gcn_bias_softmax_cls<<<clsBlocks, 256, 0, stream>>>(bufD, b2, (float*)d_out, N);
}